// Kohonen_61005715472446
// MI455X (gfx1250) — compile-verified
//
#include <hip/hip_runtime.h>

// MI455X (gfx1250) implementation: fp32 implicit-GEMM convs on the WMMA pipe
// (V_WMMA_F32_16X16X4_F32), batch-norm via global-atomic stats + fold into
// scale/shift, wave32 shuffle reductions.

typedef float v2f __attribute__((ext_vector_type(2)));
typedef float v8f __attribute__((ext_vector_type(8)));

#define NSAMP 2048

static __device__ __forceinline__ float lrelu(float x) {
    return x > 0.0f ? x : 0.01f * x;
}

// ---------------------------------------------------------------- utilities
__global__ void zero_kernel(float* p, int n) {
    int i = blockIdx.x * blockDim.x + threadIdx.x;
    if (i < n) p[i] = 0.0f;
}

// scale = gamma * rsqrt(var+eps); shift = beta - mean*scale
__global__ void stats_kernel(const float* __restrict__ sum, const float* __restrict__ sumsq,
                             const float* __restrict__ gamma, const float* __restrict__ beta,
                             float* __restrict__ scale, float* __restrict__ shift,
                             int nchan, float inv_count) {
    int c = blockIdx.x * blockDim.x + threadIdx.x;
    if (c < nchan) {
        float m   = sum[c] * inv_count;
        float var = sumsq[c] * inv_count - m * m;
        float sc  = gamma[c] * rsqrtf(var + 1e-5f);
        scale[c] = sc;
        shift[c] = beta[c] - m * sc;
    }
}

// ------------------------------------------------------------------- conv1
// in: obs slice [n][84][84]; W [32][64]; out y1 [n][32][400] = lrelu(conv+b)
__global__ __launch_bounds__(256) void conv1_kernel(
        const float* __restrict__ obs, const float* __restrict__ w,
        const float* __restrict__ bias, float* __restrict__ y1,
        float* __restrict__ gsum, float* __restrict__ gsq) {
    __shared__ float img[84 * 84];     // 28.2 KB
    __shared__ float wl[32 * 64];      //  8.0 KB
    __shared__ float ssum[32], ssq[32];

    const int n = blockIdx.x, tid = threadIdx.x;
    const float* src = obs + ((size_t)n * 4 + 3) * 7056;   // obs[:, :, -1]
    for (int i = tid; i < 7056; i += 256) img[i] = src[i];
    for (int i = tid; i < 2048; i += 256) wl[i] = w[i];
    if (tid < 32) { ssum[tid] = 0.0f; ssq[tid] = 0.0f; }
    __syncthreads();

    const int lane = tid & 31, wave = tid >> 5;
    const int half = lane >> 4, l16 = lane & 15;

    for (int t = wave; t < 50; t += 8) {        // 2 oc-tiles x 25 pos-tiles
        const int ocb = (t & 1) * 16;
        const int pb  = (t >> 1) * 16;
        const int p   = pb + l16;               // < 400 always
        const int iy0 = (p / 20) * 4, ix0 = (p % 20) * 4;
        const int m   = ocb + l16;
        v8f acc = {};
        for (int kb = 0; kb < 64; kb += 4) {
            const int klo = kb + 2 * half;      // even
            v2f a = *(const v2f*)(&wl[m * 64 + klo]);
            const int ky = klo >> 3, kx = klo & 7;   // kx even<=6 -> k,k+1 adjacent
            v2f b = *(const v2f*)(&img[(iy0 + ky) * 84 + ix0 + kx]);
            acc = __builtin_amdgcn_wmma_f32_16x16x4_f32(
                      false, a, false, b, (short)0, acc, false, false);
        }
        float* dst = y1 + (size_t)n * 12800;
#pragma unroll
        for (int r = 0; r < 8; ++r) {
            const int oc = ocb + r + 8 * half;
            float v = lrelu(acc[r] + bias[oc]);
            dst[oc * 400 + p] = v;
            float s = v, s2 = v * v;
            for (int off = 1; off < 16; off <<= 1) {
                s += __shfl_xor(s, off, 32); s2 += __shfl_xor(s2, off, 32);
            }
            if (l16 == 0) { atomicAdd(&ssum[oc], s); atomicAdd(&ssq[oc], s2); }
        }
    }
    __syncthreads();
    if (tid < 32) { atomicAdd(&gsum[tid], ssum[tid]); atomicAdd(&gsq[tid], ssq[tid]); }
}

// ------------------------------------------------------------------- conv2
// in: y1 [n][32][400] (BN folded on load); W [64][512]; out y2 [n][64][81]
__global__ __launch_bounds__(256) void conv2_kernel(
        const float* __restrict__ y1, const float* __restrict__ w,
        const float* __restrict__ bias, const float* __restrict__ scale,
        const float* __restrict__ shift, float* __restrict__ y2,
        float* __restrict__ gsum, float* __restrict__ gsq) {
    __shared__ float xl[32 * 400];     // 51.2 KB
    __shared__ float ssum[64], ssq[64];

    const int n = blockIdx.x, tid = threadIdx.x;
    const float* src = y1 + (size_t)n * 12800;
    for (int i = tid; i < 12800; i += 256) {
        int c = i / 400;
        xl[i] = src[i] * scale[c] + shift[c];
    }
    if (tid < 64) { ssum[tid] = 0.0f; ssq[tid] = 0.0f; }
    __syncthreads();

    const int lane = tid & 31, wave = tid >> 5;
    const int half = lane >> 4, l16 = lane & 15;

    for (int t = wave; t < 24; t += 8) {        // 4 oc-tiles x 6 pos-tiles
        const int ocb = (t % 4) * 16;
        const int pb  = (t / 4) * 16;
        const int p   = pb + l16;
        const bool valid = p < 81;
        const int pc  = valid ? p : 80;
        const int ib  = (pc / 9) * 40 + (pc % 9) * 2;   // stride-2 input base
        const int m   = ocb + l16;
        const float* wrow = w + m * 512;
        v8f acc = {};
        for (int kb = 0; kb < 512; kb += 4) {
            const int klo = kb + 2 * half;      // even -> c same for k,k+1
            v2f a = *(const v2f*)(wrow + klo);
            const int c0 = klo >> 4, r0 = klo & 15;     // r0 even<=14
            v2f b = *(const v2f*)(&xl[c0 * 400 + ib + (r0 >> 2) * 20 + (r0 & 3)]);
            acc = __builtin_amdgcn_wmma_f32_16x16x4_f32(
                      false, a, false, b, (short)0, acc, false, false);
        }
        float* dst = y2 + (size_t)n * 5184;
#pragma unroll
        for (int r = 0; r < 8; ++r) {
            const int oc = ocb + r + 8 * half;
            float v = lrelu(acc[r] + bias[oc]);
            float vm = valid ? v : 0.0f;
            if (valid) dst[oc * 81 + p] = v;
            float s = vm, s2 = vm * vm;
            for (int off = 1; off < 16; off <<= 1) {
                s += __shfl_xor(s, off, 32); s2 += __shfl_xor(s2, off, 32);
            }
            if (l16 == 0) { atomicAdd(&ssum[oc], s); atomicAdd(&ssq[oc], s2); }
        }
    }
    __syncthreads();
    if (tid < 64) { atomicAdd(&gsum[tid], ssum[tid]); atomicAdd(&gsq[tid], ssq[tid]); }
}

// ------------------------------------------------------------------- conv3
// in: y2 [n][64][81] (BN folded); W [64][576]; out y3 [n][64][49]
__global__ __launch_bounds__(256) void conv3_kernel(
        const float* __restrict__ y2, const float* __restrict__ w,
        const float* __restrict__ bias, const float* __restrict__ scale,
        const float* __restrict__ shift, float* __restrict__ y3,
        float* __restrict__ gsum, float* __restrict__ gsq) {
    __shared__ float xl[64 * 81];      // 20.7 KB
    __shared__ float ssum[64], ssq[64];

    const int n = blockIdx.x, tid = threadIdx.x;
    const float* src = y2 + (size_t)n * 5184;
    for (int i = tid; i < 5184; i += 256) {
        int c = i / 81;
        xl[i] = src[i] * scale[c] + shift[c];
    }
    if (tid < 64) { ssum[tid] = 0.0f; ssq[tid] = 0.0f; }
    __syncthreads();

    const int lane = tid & 31, wave = tid >> 5;
    const int half = lane >> 4, l16 = lane & 15;

    for (int t = wave; t < 16; t += 8) {        // 4 oc-tiles x 4 pos-tiles
        const int ocb = (t % 4) * 16;
        const int pb  = (t / 4) * 16;
        const int p   = pb + l16;
        const bool valid = p < 49;
        const int pc  = valid ? p : 48;
        const int ib  = (pc / 7) * 9 + (pc % 7);
        const int m   = ocb + l16;
        const float* wrow = w + m * 576;
        v8f acc = {};
        for (int kb = 0; kb < 576; kb += 4) {
            const int klo = kb + 2 * half;
            v2f a = *(const v2f*)(wrow + klo);
            const int c0 = klo / 9, r0 = klo - c0 * 9;          // k -> (c,ky,kx)
            const int k1 = klo + 1, c1 = k1 / 9, r1 = k1 - c1 * 9;
            v2f b;
            b.x = xl[c0 * 81 + ib + (r0 / 3) * 9 + (r0 % 3)];
            b.y = xl[c1 * 81 + ib + (r1 / 3) * 9 + (r1 % 3)];
            acc = __builtin_amdgcn_wmma_f32_16x16x4_f32(
                      false, a, false, b, (short)0, acc, false, false);
        }
        float* dst = y3 + (size_t)n * 3136;
#pragma unroll
        for (int r = 0; r < 8; ++r) {
            const int oc = ocb + r + 8 * half;
            float v = lrelu(acc[r] + bias[oc]);
            float vm = valid ? v : 0.0f;
            if (valid) dst[oc * 49 + p] = v;
            float s = vm, s2 = vm * vm;
            for (int off = 1; off < 16; off <<= 1) {
                s += __shfl_xor(s, off, 32); s2 += __shfl_xor(s2, off, 32);
            }
            if (l16 == 0) { atomicAdd(&ssum[oc], s); atomicAdd(&ssq[oc], s2); }
        }
    }
    __syncthreads();
    if (tid < 64) { atomicAdd(&gsum[tid], ssum[tid]); atomicAdd(&gsq[tid], ssq[tid]); }
}

// --------------------------------------------------------------------- fc
// f[n][j] = sum_i bn3(y3)[n][i] * fcw[i][j] + fcb[j]; accumulate BN-f stats
__global__ __launch_bounds__(256) void fc_kernel(
        const float* __restrict__ y3, const float* __restrict__ fcw,
        const float* __restrict__ fcb, const float* __restrict__ scale,
        const float* __restrict__ shift, float* __restrict__ f,
        float* __restrict__ gsum, float* __restrict__ gsq) {
    __shared__ float red[3][256];
    const int n = blockIdx.x, tid = threadIdx.x;
    const float* src = y3 + (size_t)n * 3136;
    float a0 = 0.0f, a1 = 0.0f, a2 = 0.0f;
    for (int i = tid; i < 3136; i += 256) {
        int c = i / 49;
        float xv = src[i] * scale[c] + shift[c];
        const float* wr = fcw + (size_t)i * 3;
        a0 += xv * wr[0]; a1 += xv * wr[1]; a2 += xv * wr[2];
    }
    red[0][tid] = a0; red[1][tid] = a1; red[2][tid] = a2;
    __syncthreads();
    for (int s = 128; s > 0; s >>= 1) {
        if (tid < s) {
            red[0][tid] += red[0][tid + s];
            red[1][tid] += red[1][tid + s];
            red[2][tid] += red[2][tid + s];
        }
        __syncthreads();
    }
    if (tid < 3) {
        float v = red[tid][0] + fcb[tid];
        f[n * 3 + tid] = v;
        atomicAdd(&gsum[tid], v);
        atomicAdd(&gsq[tid], v * v);
    }
}

// -------------------------------------------------------------------- som
__global__ __launch_bounds__(256) void som_kernel(
        const float* __restrict__ f, const float* __restrict__ scale,
        const float* __restrict__ shift, const float* __restrict__ Wsom,
        float* __restrict__ out) {
    __shared__ float Wl[300];
    const int tid = threadIdx.x;
    for (int i = tid; i < 300; i += 256) Wl[i] = Wsom[i];
    __syncthreads();
    const int n = blockIdx.x * 256 + tid;
    const float f0 = f[n * 3 + 0] * scale[0] + shift[0];
    const float f1 = f[n * 3 + 1] * scale[1] + shift[1];
    const float f2 = f[n * 3 + 2] * scale[2] + shift[2];
    int bmu = 0; float best = 3.4e38f;
    for (int k = 0; k < 100; ++k) {            // first-min matches jnp.argmin
        float d0 = f0 - Wl[k * 3], d1 = f1 - Wl[k * 3 + 1], d2 = f2 - Wl[k * 3 + 2];
        float d = d0 * d0 + d1 * d1 + d2 * d2;
        if (d < best) { best = d; bmu = k; }
    }
    const float bi = (float)(bmu / 10), bj = (float)(bmu % 10);
    const float inv2pi = 0.15915494309189535f;
    float acc = 0.0f;
    for (int k = 0; k < 100; ++k) {
        float d0 = f0 - Wl[k * 3], d1 = f1 - Wl[k * 3 + 1], d2 = f2 - Wl[k * 3 + 2];
        float dsq = d0 * d0 + d1 * d1 + d2 * d2;
        float di = (float)(k / 10) - bi, dj = (float)(k % 10) - bj;
        float g = expf(-0.5f * (di * di + dj * dj)) * inv2pi;
        acc += g * g * dsq;
    }
    out[n] = sqrtf(acc);
}

// ------------------------------------------------------------------ launch
extern "C" void kernel_launch(void* const* d_in, const int* in_sizes, int n_in,
                              void* d_out, int out_size, void* d_ws, size_t ws_size,
                              hipStream_t stream) {
    const float* obs     = (const float*)d_in[0];
    const float* conv1_w = (const float*)d_in[1];
    const float* conv1_b = (const float*)d_in[2];
    const float* bn1_g   = (const float*)d_in[3];
    const float* bn1_b   = (const float*)d_in[4];
    const float* conv2_w = (const float*)d_in[5];
    const float* conv2_b = (const float*)d_in[6];
    const float* bn2_g   = (const float*)d_in[7];
    const float* bn2_b   = (const float*)d_in[8];
    const float* conv3_w = (const float*)d_in[9];
    const float* conv3_b = (const float*)d_in[10];
    const float* bn3_g   = (const float*)d_in[11];
    const float* bn3_b   = (const float*)d_in[12];
    const float* fc_w    = (const float*)d_in[13];
    const float* fc_b    = (const float*)d_in[14];
    const float* bnf_g   = (const float*)d_in[15];
    const float* bnf_b   = (const float*)d_in[16];
    const float* W_som   = (const float*)d_in[17];

    float* ws = (float*)d_ws;
    float* y1 = ws;                          // 2048*32*400 = 26,214,400 floats
    float* y3 = ws;                          // reuse y1 region (y1 dead by then)
    float* y2 = ws + 26214400;               // 2048*64*81  = 10,616,832
    float* f  = ws + 36831232;               // 2048*3
    float* st = ws + 36837376;               // stats block (652 floats)

    float *sum1 = st,       *sq1 = st + 32,  *scale1 = st + 64,  *shift1 = st + 96;
    float *sum2 = st + 128, *sq2 = st + 192, *scale2 = st + 256, *shift2 = st + 320;
    float *sum3 = st + 384, *sq3 = st + 448, *scale3 = st + 512, *shift3 = st + 576;
    float *sumf = st + 640, *sqf = st + 643, *scalef = st + 646, *shiftf = st + 649;

    zero_kernel<<<3, 256, 0, stream>>>(st, 652);

    conv1_kernel<<<NSAMP, 256, 0, stream>>>(obs, conv1_w, conv1_b, y1, sum1, sq1);
    stats_kernel<<<1, 64, 0, stream>>>(sum1, sq1, bn1_g, bn1_b, scale1, shift1,
                                       32, 1.0f / 819200.0f);
    conv2_kernel<<<NSAMP, 256, 0, stream>>>(y1, conv2_w, conv2_b, scale1, shift1,
                                            y2, sum2, sq2);
    stats_kernel<<<1, 64, 0, stream>>>(sum2, sq2, bn2_g, bn2_b, scale2, shift2,
                                       64, 1.0f / 165888.0f);
    conv3_kernel<<<NSAMP, 256, 0, stream>>>(y2, conv3_w, conv3_b, scale2, shift2,
                                            y3, sum3, sq3);
    stats_kernel<<<1, 64, 0, stream>>>(sum3, sq3, bn3_g, bn3_b, scale3, shift3,
                                       64, 1.0f / 100352.0f);
    fc_kernel<<<NSAMP, 256, 0, stream>>>(y3, fc_w, fc_b, scale3, shift3,
                                         f, sumf, sqf);
    stats_kernel<<<1, 64, 0, stream>>>(sumf, sqf, bnf_g, bnf_b, scalef, shiftf,
                                       3, 1.0f / 2048.0f);
    som_kernel<<<8, 256, 0, stream>>>(f, scalef, shiftf, W_som, (float*)d_out);
}